// FuN_5566277616032
// MI455X (gfx1250) — compile-verified
//
#include <hip/hip_runtime.h>
#include <math.h>

typedef __attribute__((ext_vector_type(2))) float v2f;
typedef __attribute__((ext_vector_type(8))) float v8f;

#define DD 2048
#define RR 10
#define HH 1152
#define NAA 18
#define KK 64

__device__ __forceinline__ float sigm(float x) { return 1.0f / (1.0f + expf(-x)); }

// ---------------------------------------------------------------------------
// WMMA f32 matvec: out[row] = dot(W1[row,:k1], v1) [+ dot(W2[row,:k2], v2)]
//                             [+ b1[row]] [+ b2[row]]  (optional relu)
// One wave computes 16 rows using V_WMMA_F32_16X16X4_F32:
//   A tile = 16x4 weights  (lane m & lane m+16 hold K=2h..2h+1 of row m)
//   B tile = 4x16 with every column = the vector chunk  -> all 16 columns of
//   the f32 accumulator hold the same partial dot; K-loop accumulates in C.
// Result rows 0..7 live in lane 0's acc[0..7], rows 8..15 in lane 16's.
// ---------------------------------------------------------------------------
__device__ __forceinline__ v8f wmma_dot16(const float* __restrict__ W, int ldw,
                                          const float* __restrict__ vec,
                                          int kdim, int row0, int m, int h, v8f acc) {
    const float* wr = W + (size_t)(row0 + m) * (size_t)ldw + 2 * h;
    const float* vk = vec + 2 * h;
#pragma unroll 4
    for (int k = 0; k < kdim; k += 4) {
        v2f a, b;
        a[0] = wr[k];
        a[1] = wr[k + 1];
        b[0] = vk[k];
        b[1] = vk[k + 1];
        acc = __builtin_amdgcn_wmma_f32_16x16x4_f32(false, a, false, b, (short)0, acc,
                                                    false, false);
    }
    return acc;
}

__global__ __launch_bounds__(256) void wmma_matvec2(
    const float* __restrict__ W1, const float* __restrict__ v1, int k1,
    const float* __restrict__ W2, const float* __restrict__ v2b, int k2, int v2stride,
    const int* __restrict__ tickp, const float* __restrict__ b1,
    const float* __restrict__ b2, float* __restrict__ out, int rows, int do_relu) {
    int wid = blockIdx.x * (blockDim.x >> 5) + (threadIdx.x >> 5);
    int row0 = wid * 16;
    if (row0 >= rows) return;  // wave-uniform: EXEC stays all-ones for WMMA
    int lane = threadIdx.x & 31;
    int m = lane & 15;
    int h = lane >> 4;

    v8f acc = {};
    acc = wmma_dot16(W1, k1, v1, k1, row0, m, h, acc);
    if (W2) {
        const float* v2 = v2b + (tickp ? (size_t)(*tickp) * (size_t)v2stride : 0);
        acc = wmma_dot16(W2, k2, v2, k2, row0, m, h, acc);
    }

    if (m == 0) {  // lane 0 -> rows row0..+7, lane 16 -> rows row0+8..+15
        int base = row0 + (h ? 8 : 0);
#pragma unroll
        for (int r = 0; r < 8; r++) {
            int row = base + r;
            if (row < rows) {
                float val = acc[r];
                if (b1) val += b1[row];
                if (b2) val += b2[row];
                if (do_relu) val = fmaxf(val, 0.0f);
                out[row] = val;
            }
        }
    }
}

// z = relu(Wp @ x + bp),  Wp is (D,2)
__global__ void percept_kernel(const float* __restrict__ x, const float* __restrict__ Wp,
                               const float* __restrict__ bp, float* __restrict__ z) {
    int i = blockIdx.x * blockDim.x + threadIdx.x;
    if (i < DD) z[i] = fmaxf(Wp[2 * i] * x[0] + Wp[2 * i + 1] * x[1] + bp[i], 0.0f);
}

// Manager cell + ring-buffer mean + normalize + m_value + goals_sum
__global__ __launch_bounds__(256) void manager_post(
    const float* __restrict__ gm, const float* __restrict__ cn_m,
    const float* __restrict__ hn_m, const float* __restrict__ goals,
    const float* __restrict__ Wmv, const float* __restrict__ bmv,
    const int* __restrict__ tickp, float* __restrict__ g, float* __restrict__ goals_sum,
    float* __restrict__ out) {
    __shared__ float red[256];
    __shared__ float red2[256];
    __shared__ float s_inv;
    int tid = threadIdx.x;
    int tick = *tickp;
    float nloc = 0.0f, mvloc = 0.0f;
    for (int j = tid; j < DD; j += 256) {
        float gi = gm[j], gf = gm[DD + j], gg = gm[2 * DD + j], go = gm[3 * DD + j];
        float c2 = sigm(gf) * cn_m[(size_t)tick * DD + j] + sigm(gi) * tanhf(gg);
        float hnew = sigm(go) * tanhf(c2);
        float gh = hnew;
        for (int r = 0; r < RR; r++)
            if (r != tick) gh += hn_m[(size_t)r * DD + j];
        gh *= (1.0f / (float)RR);
        g[j] = gh;  // temporarily g_hat
        nloc += gh * gh;
        mvloc += Wmv[j] * gh;
    }
    red[tid] = nloc;
    red2[tid] = mvloc;
    __syncthreads();
    for (int s = 128; s > 0; s >>= 1) {
        if (tid < s) {
            red[tid] += red[tid + s];
            red2[tid] += red2[tid + s];
        }
        __syncthreads();
    }
    if (tid == 0) {
        out[20] = red2[0] + bmv[0];                       // m_value
        s_inv = 1.0f / fmaxf(sqrtf(red[0]), 1e-12f);      // 1/||g_hat||
    }
    __syncthreads();
    for (int j = tid; j < DD; j += 256) {
        float gv = g[j] * s_inv;
        g[j] = gv;  // normalized goal
        float gs = gv;
        for (int r = 1; r < RR; r++) gs += goals[(size_t)r * DD + j];
        goals_sum[j] = gs;  // sum of goals2 rows
    }
}

// Worker LSTM cell: u = sigmoid(o) * tanh(sigmoid(f)*c + sigmoid(i)*tanh(g))
__global__ void worker_cell(const float* __restrict__ gw, const float* __restrict__ cn_w,
                            float* __restrict__ u) {
    int j = blockIdx.x * blockDim.x + threadIdx.x;
    if (j < HH) {
        float gi = gw[j], gf = gw[HH + j], gg = gw[2 * HH + j], go = gw[3 * HH + j];
        float c2 = sigm(gf) * cn_w[j] + sigm(gi) * tanhf(gg);
        u[j] = sigm(go) * tanhf(c2);
    }
}

// logits, w_value, intrinsic reward
__global__ __launch_bounds__(256) void final_kernel(
    const float* __restrict__ u, const float* __restrict__ w,
    const float* __restrict__ Wc, const float* __restrict__ bc,
    const float* __restrict__ s, const float* __restrict__ states,
    const float* __restrict__ goals, const float* __restrict__ g,
    float* __restrict__ out) {
    __shared__ float r0[256], r1[256], r2[256];
    int tid = threadIdx.x;

    // w_value = Wc @ u + bc
    float acc = 0.0f;
    for (int j = tid; j < HH; j += 256) acc += Wc[j] * u[j];
    r0[tid] = acc;
    __syncthreads();
    for (int sN = 128; sN > 0; sN >>= 1) {
        if (tid < sN) r0[tid] += r0[tid + sN];
        __syncthreads();
    }
    if (tid == 0) out[19] = r0[0] + bc[0];

    // logits = u.reshape(NA,K) @ w
    if (tid < NAA) {
        float l = 0.0f;
        for (int kk = 0; kk < KK; kk++) l += u[tid * KK + kk] * w[kk];
        out[tid] = l;
    }
    __syncthreads();

    // intrinsic reward: cos(s - states[1+r], goals2[r]) summed, scaled by D/C
    float cs = 0.0f;
    for (int r = 0; r < RR; r++) {
        float a = 0.0f, bq = 0.0f, cq = 0.0f;
        for (int j = tid; j < DD; j += 256) {
            float d = s[j] - states[(size_t)(1 + r) * DD + j];
            float gl = (r < RR - 1) ? goals[(size_t)(r + 1) * DD + j] : g[j];
            a += d * gl;
            bq += d * d;
            cq += gl * gl;
        }
        r0[tid] = a;
        r1[tid] = bq;
        r2[tid] = cq;
        __syncthreads();
        for (int sN = 128; sN > 0; sN >>= 1) {
            if (tid < sN) {
                r0[tid] += r0[tid + sN];
                r1[tid] += r1[tid + sN];
                r2[tid] += r2[tid + sN];
            }
            __syncthreads();
        }
        if (tid == 0) {
            float den = fmaxf(sqrtf(r1[0]), 1e-8f) * fmaxf(sqrtf(r2[0]), 1e-8f);
            cs += r0[0] / den;
        }
        __syncthreads();
    }
    if (tid == 0) out[18] = (float)DD * cs / 10.0f;  // C = 10
}

extern "C" void kernel_launch(void* const* d_in, const int* in_sizes, int n_in,
                              void* d_out, int out_size, void* d_ws, size_t ws_size,
                              hipStream_t stream) {
    const float* x     = (const float*)d_in[0];
    const float* Wp    = (const float*)d_in[1];
    const float* bp    = (const float*)d_in[2];
    const float* Wms   = (const float*)d_in[3];
    const float* bms   = (const float*)d_in[4];
    const float* Wih_m = (const float*)d_in[5];
    const float* Whh_m = (const float*)d_in[6];
    const float* bih_m = (const float*)d_in[7];
    const float* bhh_m = (const float*)d_in[8];
    const float* hn_m  = (const float*)d_in[9];
    const float* cn_m  = (const float*)d_in[10];
    const float* Wmv   = (const float*)d_in[11];
    const float* bmv   = (const float*)d_in[12];
    const float* Wih_w = (const float*)d_in[13];
    const float* Whh_w = (const float*)d_in[14];
    const float* bih_w = (const float*)d_in[15];
    const float* bhh_w = (const float*)d_in[16];
    const float* hn_w  = (const float*)d_in[17];
    const float* cn_w  = (const float*)d_in[18];
    const float* Wphi  = (const float*)d_in[19];
    const float* Wc    = (const float*)d_in[20];
    const float* bc    = (const float*)d_in[21];
    const float* states= (const float*)d_in[22];
    const float* goals = (const float*)d_in[23];
    const int*   tick  = (const int*)d_in[24];
    float* out = (float*)d_out;

    float* ws   = (float*)d_ws;
    float* z    = ws;            // 2048
    float* s    = ws + 2048;     // 2048
    float* gm   = ws + 4096;     // 8192 (manager gates)
    float* g    = ws + 12288;    // 2048 (g_hat then g)
    float* gsum = ws + 14336;    // 2048 (goals2 row-sum)
    float* wv   = ws + 16384;    // 64   (w = Wphi @ goals_sum)
    float* gw   = ws + 16448;    // 4608 (worker gates)
    float* u    = ws + 21056;    // 1152

    // 1) percept
    percept_kernel<<<8, 256, 0, stream>>>(x, Wp, bp, z);
    // 2) s = relu(Wms @ z + bms)        (2048 rows)
    wmma_matvec2<<<16, 256, 0, stream>>>(Wms, z, DD, nullptr, nullptr, 0, 0, nullptr,
                                         bms, nullptr, s, DD, 1);
    // 3) worker gates = Wih_w@z + Whh_w@hn_w + biases   (4608 rows)
    wmma_matvec2<<<36, 256, 0, stream>>>(Wih_w, z, DD, Whh_w, hn_w, HH, 0, nullptr,
                                         bih_w, bhh_w, gw, 4 * HH, 0);
    // 4) manager gates = Wih_m@s + Whh_m@hn_m[tick] + biases   (8192 rows)
    wmma_matvec2<<<64, 256, 0, stream>>>(Wih_m, s, DD, Whh_m, hn_m, DD, DD, tick,
                                         bih_m, bhh_m, gm, 4 * DD, 0);
    // 5) manager cell, g, m_value, goals_sum
    manager_post<<<1, 256, 0, stream>>>(gm, cn_m, hn_m, goals, Wmv, bmv, tick, g, gsum,
                                        out);
    // 6) w = Wphi @ goals_sum   (64 rows)
    wmma_matvec2<<<1, 128, 0, stream>>>(Wphi, gsum, DD, nullptr, nullptr, 0, 0, nullptr,
                                        nullptr, nullptr, wv, KK, 0);
    // 7) worker cell -> u
    worker_cell<<<5, 256, 0, stream>>>(gw, cn_w, u);
    // 8) logits, w_value, intrinsic reward
    final_kernel<<<1, 256, 0, stream>>>(u, wv, Wc, bc, s, states, goals, g, out);
}